// Input3d_50019189129828
// MI455X (gfx1250) — compile-verified
//
#include <hip/hip_runtime.h>

typedef __attribute__((ext_vector_type(2))) float v2f;
typedef __attribute__((ext_vector_type(4))) float v4f;
typedef __attribute__((ext_vector_type(8))) float v8f;

#define BB      4
#define NV      50000
#define NRINGS  3
#define NDIRS   16
#define NF      16
#define KTOT    (NRINGS * 3 * NDIRS)   // 144, K order: k = (r*3+c)*16 + dw
#define WPB     8                      // waves per block (256 threads, wave32)
#define YPAD    31                     // circularly padded direction axis (16 + 15)
#define YSTRIDE 288                    // >= 9*YPAD (279) and >= 256 (out staging)

// ---------------------------------------------------------------------------
// Pre-pass: permute the (3,16,3,NF) conv kernel into the pair-interleaved
// B-matrix layout   kperm[(kn>>1)*32 + f*2 + (kn&1)] = B[kn][f],
// with K order kn = (r*3+c)*16 + dw.  Runs once per launch (9 blocks);
// removes all index-decomposition VALU from the 25000-block main kernel.
// ---------------------------------------------------------------------------
__global__ void permute_kernel_tile(const float* __restrict__ kernel,
                                    float* __restrict__ kperm) {
    const int j = blockIdx.x * blockDim.x + threadIdx.x;   // destination index
    if (j >= KTOT * NF) return;
    const int f  = (j & 31) >> 1;
    const int kn = ((j >> 5) << 1) | (j & 1);
    const int rc = kn >> 4;            // r*3 + c
    const int dw = kn & 15;
    const int r  = rc / 3;
    const int c  = rc - r * 3;
    kperm[j] = kernel[((r * NDIRS + dw) * 3 + c) * NF + f];
}

// ---------------------------------------------------------------------------
// Main kernel: one wave32 per vertex; per-vertex 16x16x144 f32 GEMM on the
// matrix pipe (36 chained V_WMMA_F32_16X16X4_F32).
// ---------------------------------------------------------------------------
__global__ __launch_bounds__(256, 2)
void geodesic_patch_conv_wmma(const float* __restrict__ basepoints,   // (B,NV,3)
                              const float* __restrict__ frames,       // (B,NV,3,3)
                              const int*   __restrict__ expmap,       // (B,NV,3,16,2) int32
                              const float* __restrict__ kperm,        // pre-permuted B tile
                              const float* __restrict__ bias,         // (NF)
                              float*       __restrict__ out)          // (B,NV,16,NF)
{
    __shared__ __align__(16) float kbt[KTOT * NF];          // 9216 B (B tile)
    __shared__ __align__(16) float ybuf[WPB][YSTRIDE];      // 9216 B (patch / out staging)

    const int tid  = threadIdx.x;
    const int lane = tid & 31;
    const int wave = tid >> 5;
    const int half = lane >> 4;            // 0: lanes 0-15, 1: lanes 16-31
    const int wn   = lane & 15;            // A row (M) / B column (N) within half

    // ---- stage pre-permuted B tile: plain b128 copy, no index math ----
    {
        const v4f* src = (const v4f*)kperm;
        v4f*       dst = (v4f*)kbt;
#pragma unroll
        for (int i = 0; i < (KTOT * NF) / 4; i += 256) {
            const int idx = i + tid;
            if (idx < (KTOT * NF) / 4) dst[idx] = src[idx];
        }
    }

    // ---- this wave's vertex ----
    const int g = blockIdx.x * WPB + wave;     // 0 .. B*NV-1 (exact fit)
    const int b = g / NV;

    const float* cp = basepoints + (size_t)g * 3;
    const float c0 = cp[0], c1 = cp[1], c2 = cp[2];
    const float* F = frames + (size_t)g * 9;
    const float F00=F[0],F01=F[1],F02=F[2],
                F10=F[3],F11=F[4],F12=F[5],
                F20=F[6],F21=F[7],F22=F[8];

    // ---- gather + rotate -> circularly padded LDS patch ypad[rc*31 + t] ----
    float* yl = ybuf[wave];
    for (int p = lane; p < NRINGS * NDIRS; p += 32) {     // 48 (ring,dir) pairs
        const int r = p >> 4;
        const int d = p & 15;
        const int nidx = expmap[(((size_t)g * NRINGS + r) * NDIRS + d) * 2 + 1];
        const float* np = basepoints + ((size_t)b * NV + nidx) * 3;
        const float dx = np[0] - c0, dy = np[1] - c1, dz = np[2] - c2;
        const float v0 = F00*dx + F01*dy + F02*dz;
        const float v1 = F10*dx + F11*dy + F12*dz;
        const float v2 = F20*dx + F21*dy + F22*dz;
        const int b0 = (r * 3 + 0) * YPAD + d;
        const int b1 = (r * 3 + 1) * YPAD + d;
        const int b2 = (r * 3 + 2) * YPAD + d;
        yl[b0] = v0;  yl[b1] = v1;  yl[b2] = v2;
        if (d < NDIRS - 1) {                   // circular padding t = 16..30
            yl[b0 + NDIRS] = v0;  yl[b1 + NDIRS] = v1;  yl[b2 + NDIRS] = v2;
        }
    }

    __syncthreads();   // covers B tile + patch staging

    // ---- 16x16x144 GEMM: 36 chained V_WMMA_F32_16X16X4_F32 ----
    // Iteration kk handles K block k0 = 4*kk (+2 for upper half-wave):
    //   A addr  = yl[(kk>>2)*31 + (kk&3)*4 + (wn + 2*half)]   (+1 for a.y)
    //   B addr  = kbt[64*kk + half*32 + wn*2]                 (aligned b64)
    // -> all per-iteration offsets are compile-time immediates; zero VALU.
    const float* pA = yl + wn + (half << 1);
    const float* pB = kbt + (half << 5) + (wn << 1);

    v8f acc = {};
#pragma unroll
    for (int kk = 0; kk < KTOT / 4; ++kk) {
        const int oA = (kk >> 2) * YPAD + (kk & 3) * 4;
        v2f a;
        a.x = pA[oA];
        a.y = pA[oA + 1];
        const v2f bf = *(const v2f*)(pB + kk * 64);
        acc = __builtin_amdgcn_wmma_f32_16x16x4_f32(
                  /*neg_a=*/false, a, /*neg_b=*/false, bf,
                  /*c_mod=*/(short)0, acc, /*reuse_a=*/false, /*reuse_b=*/false);
    }

    // ---- epilogue: bias + ReLU, stage tile in LDS, 2x coalesced b128 stores ----
    const float bs = bias[wn];
#pragma unroll
    for (int j = 0; j < 8; ++j) {
        const int m = j + half * 8;            // output direction w
        float val = acc[j] + bs;
        val = val > 0.0f ? val : 0.0f;
        yl[m * NF + wn] = val;                 // tile layout: [m*16 + f]
    }

    // wave-private buffer: wait for all 32 lanes' ds_stores, block reordering
    asm volatile("s_wait_dscnt 0x0" ::: "memory");

    float* op = out + (size_t)g * (NDIRS * NF);          // 256 floats / vertex
    const v4f t0 = *(const v4f*)&yl[lane * 4];           // floats   0..127
    const v4f t1 = *(const v4f*)&yl[128 + lane * 4];     // floats 128..255
    *(v4f*)(op + lane * 4)       = t0;                   // 512B contiguous / wave
    *(v4f*)(op + 128 + lane * 4) = t1;
}

extern "C" void kernel_launch(void* const* d_in, const int* in_sizes, int n_in,
                              void* d_out, int out_size, void* d_ws, size_t ws_size,
                              hipStream_t stream) {
    (void)in_sizes; (void)n_in; (void)ws_size; (void)out_size;
    const float* basepoints = (const float*)d_in[0];
    const float* frames     = (const float*)d_in[1];
    const int*   expmap     = (const int*)  d_in[2];
    const float* kernel     = (const float*)d_in[3];
    const float* bias       = (const float*)d_in[4];
    float*       out        = (float*)d_out;
    float*       kperm      = (float*)d_ws;              // 9216 B scratch

    // one-shot B-tile permutation (same result every call; graph-capture safe)
    permute_kernel_tile<<<(KTOT * NF + 255) / 256, 256, 0, stream>>>(kernel, kperm);

    const int total_waves = BB * NV;              // 200000
    const int blocks = total_waves / WPB;         // 25000 (exact)
    geodesic_patch_conv_wmma<<<blocks, 256, 0, stream>>>(
        basepoints, frames, expmap, kperm, bias, out);
}